// DGMG_46600395162232
// MI455X (gfx1250) — compile-verified
//
#include <hip/hip_runtime.h>
#include <math.h>

// DGMG loss on MI455X: single persistent workgroup (32 waves on one WGP),
// state in LDS, fp32 WMMA (V_WMMA_F32_16X16X4_F32) for all big matmuls.

#define NMAX 128
#define DH   256
#define DG   512
#define TROUNDS 2
#define PH   260          // hV row pitch (floats): 260%64==4 -> conflict-free A-operand fetch
#define PA   132          // adjacency row pitch (bytes, u8 storage)
#define NTHREADS 1024
#define NWAVES   32

typedef __attribute__((ext_vector_type(2))) float v2f;
typedef __attribute__((ext_vector_type(8))) float v8f;

__device__ __forceinline__ v8f wmma_f32(v2f a, v2f b, v8f c) {
  // D = A(16x4,f32) * B(4x16,f32) + C(16x16,f32)
  return __builtin_amdgcn_wmma_f32_16x16x4_f32(
      /*neg_a=*/false, a, /*neg_b=*/false, b,
      /*c_mod=*/(short)0, c, /*reuse_a=*/false, /*reuse_b=*/false);
}

__device__ __forceinline__ float softplusf(float x) {
  // log(1+exp(x)), stable
  return fmaxf(x, 0.0f) + log1pf(expf(-fabsf(x)));
}
__device__ __forceinline__ float sigmoidf(float x) {
  return 1.0f / (1.0f + expf(-x));
}

struct SMEM {
  float hV[NMAX * PH];          // node embeddings (pitched)
  float mask[NMAX];
  float deg[NMAX];              // deg of A+diag(mask)
  float xsum[DH];               // mask^T hV
  float hG[DG];                 // graph embedding
  float hv[DH];                 // newest node embedding
  float h1[DG + DH];            // MLP hidden buffer (<=768)
  float s[NMAX];                // destination scores
  float red[NTHREADS];          // reduction scratch
  float loss;
  int   n;
  unsigned char A[NMAX * PA];   // adjacency incl. self-loop diag (u8)
};                              // ~159 KB

__device__ float block_reduce_sum(float v, float* red, int tid) {
  red[tid] = v; __syncthreads();
  for (int off = NTHREADS / 2; off > 0; off >>= 1) {
    if (tid < off) red[tid] += red[tid + off];
    __syncthreads();
  }
  float r = red[0]; __syncthreads();
  return r;
}
__device__ float block_reduce_max(float v, float* red, int tid) {
  red[tid] = v; __syncthreads();
  for (int off = NTHREADS / 2; off > 0; off >>= 1) {
    if (tid < off) red[tid] = fmaxf(red[tid], red[tid + off]);
    __syncthreads();
  }
  float r = red[0]; __syncthreads();
  return r;
}

__global__ __launch_bounds__(NTHREADS, 1)
void dgmg_kernel(const int* __restrict__ step_type, const int* __restrict__ edge_dst,
                 const float* __restrict__ Wgcn, const float* __restrict__ bgcn,
                 const float* __restrict__ Wp,   const float* __restrict__ bp,
                 const float* __restrict__ fan_W1, const float* __restrict__ fan_b1,
                 const float* __restrict__ fan_W2, const float* __restrict__ fan_b2,
                 const float* __restrict__ fae_W1, const float* __restrict__ fae_b1,
                 const float* __restrict__ fae_W2, const float* __restrict__ fae_b2,
                 const float* __restrict__ fs_W1,  const float* __restrict__ fs_b1,
                 const float* __restrict__ fs_W2,  const float* __restrict__ fs_b2,
                 const float* __restrict__ fi_W1,  const float* __restrict__ fi_b1,
                 const float* __restrict__ fi_W2,  const float* __restrict__ fi_b2,
                 float* __restrict__ Gws,   // [128*256] GCN intermediate in global ws
                 float* __restrict__ out, int S)
{
  __shared__ SMEM sm;
  const int tid  = threadIdx.x;
  const int lane = tid & 31;
  const int w    = tid >> 5;          // wave id 0..31
  const int ln   = lane & 15;         // row/col within tile
  const int hs   = lane >> 4;         // half-wave select
  const int kb   = hs * 2;            // k sub-offset per ISA A/B layout

  // ---- init state ----
  for (int i = tid; i < NMAX * PH; i += NTHREADS) sm.hV[i] = 0.0f;
  for (int i = tid; i < NMAX * PA; i += NTHREADS) sm.A[i]  = 0;
  for (int i = tid; i < NMAX; i += NTHREADS) { sm.mask[i] = 0.0f; sm.deg[i] = 0.0f; sm.s[i] = 0.0f; }
  if (tid == 0) { sm.loss = 0.0f; sm.n = 0; }
  __syncthreads();

  for (int step = 0; step <= S; ++step) {
    const int st  = (step < S) ? step_type[step] : 3;   // 3 = final stop-node
    const int dst = (step < S) ? edge_dst[step]  : 0;
    const int n   = sm.n;

    // ---- hG = (mask^T hV) @ Wp + n*bp  (gemb factored) ----
    if (tid < DH) {
      float acc = 0.0f;
      for (int v = 0; v < NMAX; ++v) acc += sm.mask[v] * sm.hV[v * PH + tid];
      sm.xsum[tid] = acc;
    }
    __syncthreads();
    if (tid < DG) {
      float acc = (float)n * bp[tid];
      for (int k = 0; k < DH; ++k) acc += sm.xsum[k] * Wp[k * DG + tid];
      sm.hG[tid] = acc;
    }
    __syncthreads();

    if (st == 0) {
      // ================= add_node =================
      if (tid < DG) {
        float a = fan_b1[tid];
        for (int k = 0; k < DG; ++k) a += sm.hG[k] * fan_W1[k * DG + tid];
        sm.h1[tid] = sigmoidf(a);
      }
      __syncthreads();
      float p0 = 0.0f, p1 = 0.0f;
      if (tid < DG) { float h = sm.h1[tid]; p0 = h * fan_W2[tid * 2 + 0]; p1 = h * fan_W2[tid * 2 + 1]; }
      float z0 = block_reduce_sum(p0, sm.red, tid) + fan_b2[0];
      float z1 = block_reduce_sum(p1, sm.red, tid) + fan_b2[1];
      if (tid == 0) {
        float m = fmaxf(z0, z1);
        float lse = m + logf(expf(z0 - m) + expf(z1 - m));
        sm.loss += lse - z1;                 // -log_softmax[1]
      }
      // fi MLP -> new node embedding
      if (tid < DG) {
        float a = fi_b1[tid];
        for (int k = 0; k < DG; ++k) a += sm.hG[k] * fi_W1[k * DG + tid];
        sm.h1[tid] = sigmoidf(a);
      }
      __syncthreads();
      if (tid < DH) {
        float a = fi_b2[tid];
        for (int k = 0; k < DG; ++k) a += sm.h1[k] * fi_W2[k * DH + tid];
        sm.hV[n * PH + tid] = a;
      }
      if (tid == 0) {
        sm.mask[n] = 1.0f;
        sm.deg[n]  = 1.0f;                   // self-loop from diag(mask)
        sm.A[n * PA + n] = 1;
        sm.n = n + 1;
      }
    } else if (st == 1 || st == 2) {
      // ============ add_edge / stop_edge ============
      const int nm1 = n - 1;
      if (tid < DH) sm.hv[tid] = sm.hV[nm1 * PH + tid];
      __syncthreads();
      // fae MLP on concat(hG, hv) (768)
      if (tid < DG + DH) {
        float a = fae_b1[tid];
        for (int k = 0; k < DG; ++k) a += sm.hG[k] * fae_W1[k * (DG + DH) + tid];
        for (int k = 0; k < DH; ++k) a += sm.hv[k] * fae_W1[(DG + k) * (DG + DH) + tid];
        sm.h1[tid] = sigmoidf(a);
      }
      __syncthreads();
      float p = (tid < DG + DH) ? sm.h1[tid] * fae_W2[tid] : 0.0f;
      float logit = block_reduce_sum(p, sm.red, tid) + fae_b2[0];
      if (tid == 0) sm.loss += softplusf(st == 1 ? -logit : logit);

      if (st == 1) {
        // ---- fs scores: s = fs_W2 . sigmoid([hV|hv] @ fs_W1 + b1), fused WMMA ----
        if (tid < NMAX) sm.s[tid] = fs_b2[0];
        __syncthreads();
        {
          const int nt = w;                               // wave owns one 16-col strip
          const float b1c = fs_b1[nt * 16 + ln];
          const float w2c = fs_W2[nt * 16 + ln];
          for (int mt = 0; mt < NMAX / 16; ++mt) {
            v8f c = {};
            const int m = mt * 16 + ln;
            for (int ks = 0; ks < DG / 4; ++ks) {
              const int k0 = ks * 4 + kb;
              v2f a, b;
              if (ks * 4 < DH) { a.x = sm.hV[m * PH + k0]; a.y = sm.hV[m * PH + k0 + 1]; }
              else             { a.x = sm.hv[k0 - DH];     a.y = sm.hv[k0 - DH + 1]; }
              b.x = fs_W1[k0 * DG + nt * 16 + ln];
              b.y = fs_W1[(k0 + 1) * DG + nt * 16 + ln];
              c = wmma_f32(a, b, c);
            }
            float pr[8];
#pragma unroll
            for (int r = 0; r < 8; ++r) {
              float q = sigmoidf(c[r] + b1c) * w2c;
              q += __shfl_xor(q, 1, 32);
              q += __shfl_xor(q, 2, 32);
              q += __shfl_xor(q, 4, 32);
              q += __shfl_xor(q, 8, 32);
              pr[r] = q;                                   // sum over 16 cols of this strip
            }
            if (ln == 0) {
#pragma unroll
              for (int r = 0; r < 8; ++r)
                atomicAdd(&sm.s[mt * 16 + hs * 8 + r], pr[r]);
            }
          }
        }
        __syncthreads();
        // masked log_softmax over first n-1 nodes
        float sv = (tid < NMAX) ? ((tid < nm1) ? sm.s[tid] : -1e9f) : -3e38f;
        float smax = block_reduce_max(sv, sm.red, tid);
        float se = (tid < nm1) ? expf(sm.s[tid] - smax) : 0.0f;
        float ssum = block_reduce_sum(se, sm.red, tid);
        if (tid == 0) {
          sm.loss += (smax + logf(ssum)) - sm.s[dst];
          sm.A[nm1 * PA + dst] = 1;
          sm.A[dst * PA + nm1] = 1;
          sm.deg[nm1] += 1.0f;
          sm.deg[dst] += 1.0f;
        }
        __syncthreads();

        // ---- GCN: H = relu(inv .* (Ahat @ (H @ Wgcn[t])) + b) .* mask, T rounds ----
        for (int t = 0; t < TROUNDS; ++t) {
          const float* Wt = Wgcn + t * DH * DH;
          const float* bt = bgcn + t * DH;
          // G = hV @ Wt  -> global ws (128x256), 8x16 tiles, 4 per wave
          for (int tt = w; tt < (NMAX / 16) * (DH / 16); tt += NWAVES) {
            const int mt = tt >> 4, nt2 = tt & 15;
            const int m = mt * 16 + ln;
            v8f c = {};
            for (int ks = 0; ks < DH / 4; ++ks) {
              const int k0 = ks * 4 + kb;
              v2f a, b;
              a.x = sm.hV[m * PH + k0];
              a.y = sm.hV[m * PH + k0 + 1];
              b.x = Wt[k0 * DH + nt2 * 16 + ln];
              b.y = Wt[(k0 + 1) * DH + nt2 * 16 + ln];
              c = wmma_f32(a, b, c);
            }
#pragma unroll
            for (int r = 0; r < 8; ++r)
              Gws[(mt * 16 + hs * 8 + r) * DH + nt2 * 16 + ln] = c[r];
          }
          __threadfence();
          __syncthreads();
          // hV = relu(inv * (Ahat @ G) + b) * mask
          for (int tt = w; tt < (NMAX / 16) * (DH / 16); tt += NWAVES) {
            const int mt = tt >> 4, nt2 = tt & 15;
            const int m = mt * 16 + ln;
            v8f c = {};
            for (int ks = 0; ks < NMAX / 4; ++ks) {
              const int k0 = ks * 4 + kb;
              v2f a, b;
              a.x = (float)sm.A[m * PA + k0];
              a.y = (float)sm.A[m * PA + k0 + 1];
              b.x = Gws[k0 * DH + nt2 * 16 + ln];
              b.y = Gws[(k0 + 1) * DH + nt2 * 16 + ln];
              c = wmma_f32(a, b, c);
            }
            const float bc = bt[nt2 * 16 + ln];
#pragma unroll
            for (int r = 0; r < 8; ++r) {
              const int row = mt * 16 + hs * 8 + r;
              float dgr = sm.deg[row];
              float inv = (dgr > 0.0f) ? (1.0f / dgr) : 0.0f;
              sm.hV[row * PH + nt2 * 16 + ln] =
                  fmaxf(inv * c[r] + bc, 0.0f) * sm.mask[row];
            }
          }
          __syncthreads();
        }
      }
    } else {
      // ================= final stop-node (label 0) =================
      if (tid < DG) {
        float a = fan_b1[tid];
        for (int k = 0; k < DG; ++k) a += sm.hG[k] * fan_W1[k * DG + tid];
        sm.h1[tid] = sigmoidf(a);
      }
      __syncthreads();
      float p0 = 0.0f, p1 = 0.0f;
      if (tid < DG) { float h = sm.h1[tid]; p0 = h * fan_W2[tid * 2 + 0]; p1 = h * fan_W2[tid * 2 + 1]; }
      float z0 = block_reduce_sum(p0, sm.red, tid) + fan_b2[0];
      float z1 = block_reduce_sum(p1, sm.red, tid) + fan_b2[1];
      if (tid == 0) {
        float m = fmaxf(z0, z1);
        float lse = m + logf(expf(z0 - m) + expf(z1 - m));
        sm.loss += lse - z0;                 // -log_softmax[0]
        out[0] = sm.loss;
      }
    }
    __syncthreads();
  }
}

extern "C" void kernel_launch(void* const* d_in, const int* in_sizes, int n_in,
                              void* d_out, int out_size, void* d_ws, size_t ws_size,
                              hipStream_t stream) {
  const int*   step_type = (const int*)d_in[0];
  const int*   edge_dst  = (const int*)d_in[1];
  const float* Wgcn   = (const float*)d_in[2];
  const float* bgcn   = (const float*)d_in[3];
  const float* Wp     = (const float*)d_in[4];
  const float* bp     = (const float*)d_in[5];
  const float* fan_W1 = (const float*)d_in[6];
  const float* fan_b1 = (const float*)d_in[7];
  const float* fan_W2 = (const float*)d_in[8];
  const float* fan_b2 = (const float*)d_in[9];
  const float* fae_W1 = (const float*)d_in[10];
  const float* fae_b1 = (const float*)d_in[11];
  const float* fae_W2 = (const float*)d_in[12];
  const float* fae_b2 = (const float*)d_in[13];
  const float* fs_W1  = (const float*)d_in[14];
  const float* fs_b1  = (const float*)d_in[15];
  const float* fs_W2  = (const float*)d_in[16];
  const float* fs_b2  = (const float*)d_in[17];
  const float* fi_W1  = (const float*)d_in[18];
  const float* fi_b1  = (const float*)d_in[19];
  const float* fi_W2  = (const float*)d_in[20];
  const float* fi_b2  = (const float*)d_in[21];
  const int S = in_sizes[0];

  float* Gws = (float*)d_ws;   // 128*256 f32 GCN intermediate

  dgmg_kernel<<<1, NTHREADS, 0, stream>>>(
      step_type, edge_dst, Wgcn, bgcn, Wp, bp,
      fan_W1, fan_b1, fan_W2, fan_b2,
      fae_W1, fae_b1, fae_W2, fae_b2,
      fs_W1, fs_b1, fs_W2, fs_b2,
      fi_W1, fi_b1, fi_W2, fi_b2,
      Gws, (float*)d_out, S);
}